// GCN_61280593379662
// MI455X (gfx1250) — compile-verified
//
#include <hip/hip_runtime.h>
#include <hip/hip_bf16.h>

// ---------------------------------------------------------------------------
// GCN (3x GCNConv + mean pool + linear) for MI455X / gfx1250 (wave32, WMMA).
//
// Problem sizes (fixed by the reference):
//   N = 50000 nodes, F_IN = 128, E = 800000 edges, H = 64, C = 10, B = 512
//
// N = 3125*16 and H = 4*16 exactly -> WMMA tiles with no remainders, EXEC
// stays all-ones inside the GEMM kernel (required by V_WMMA_*).
// ---------------------------------------------------------------------------

#define GCN_N    50000
#define GCN_FIN  128
#define GCN_E    800000
#define GCN_H    64
#define GCN_C    10
#define GCN_B    512

typedef __attribute__((ext_vector_type(2))) float v2f;
typedef __attribute__((ext_vector_type(8))) float v8f;

// ---------------------------------------------------------------------------
// Dense GEMM  C[Nrows x 64] = A[Nrows x K] @ W[K x 64]   (row-major, fp32)
// One wave computes one 16x16 tile with V_WMMA_F32_16X16X4_F32, K-step = 4.
// Block = (32 lanes, 4 waves): 4 waves share a row tile, one col tile each.
//
// Fragment layouts (CDNA5 ISA 7.12.2):
//   A 16x4 : lanes 0-15 -> row M=lane, v0=K0, v1=K1 ; lanes 16-31 -> K2,K3
//   B 4x16 : lanes 0-15 -> col N=lane, v0=K0, v1=K1 ; lanes 16-31 -> K2,K3
//   C 16x16: VGPR v -> M = v + 8*(lane>=16), N = lane & 15
// ---------------------------------------------------------------------------
__global__ __launch_bounds__(128) void gcn_gemm_wmma_f32(
    const float* __restrict__ A, const float* __restrict__ W,
    float* __restrict__ C, int K)
{
    const int lane    = threadIdx.x;        // 0..31
    const int colBase = threadIdx.y * 16;   // 0,16,32,48
    const int rowBase = blockIdx.x * 16;
    const int half    = lane >> 4;          // 0: K0/K1 side, 1: K2/K3 side
    const int l       = lane & 15;

    const float* arow = A + (size_t)(rowBase + l) * K + 2 * half;
    const float* wcol = W + colBase + l;

    v8f acc = {};
    for (int k0 = 0; k0 < K; k0 += 4) {
        v2f a, b;
        a.x = arow[k0 + 0];                       // A[row, k0 + 2*half + 0]
        a.y = arow[k0 + 1];                       // A[row, k0 + 2*half + 1]
        b.x = wcol[(size_t)(k0 + 2 * half + 0) * GCN_H];  // W[k, col]
        b.y = wcol[(size_t)(k0 + 2 * half + 1) * GCN_H];
        acc = __builtin_amdgcn_wmma_f32_16x16x4_f32(
            /*neg_a=*/false, a, /*neg_b=*/false, b,
            /*c_mod=*/(short)0, acc, /*reuse_a=*/false, /*reuse_b=*/false);
    }

#pragma unroll
    for (int v = 0; v < 8; ++v) {
        const int m = v + 8 * half;
        C[(size_t)(rowBase + m) * GCN_H + colBase + l] = acc[v];
    }
}

// ---------------------------------------------------------------------------
// Degree count: deg[dst[e]] += 1  (deg buffer pre-zeroed)
// ---------------------------------------------------------------------------
__global__ __launch_bounds__(256) void gcn_deg_count(
    const int* __restrict__ dst, float* __restrict__ deg)
{
    const int e = blockIdx.x * blockDim.x + threadIdx.x;
    if (e < GCN_E) atomicAdd(&deg[dst[e]], 1.0f);
}

// dinv[i] = rsqrt(deg[i] + 1)   (the +1 is the self loop)
__global__ __launch_bounds__(256) void gcn_dinv(float* __restrict__ deg)
{
    const int i = blockIdx.x * blockDim.x + threadIdx.x;
    if (i < GCN_N) deg[i] = rsqrtf(deg[i] + 1.0f);
}

// ---------------------------------------------------------------------------
// Edge scatter:  agg[dst] += h[src] * dinv[src]*dinv[dst]
// 32 consecutive work-items handle one edge (one wave -> one edge),
// each lane moves a float2 (2 of the 64 features).  agg pre-zeroed.
// Whole feature table (12.8 MB) fits in L2, so atomics resolve there.
// ---------------------------------------------------------------------------
__global__ __launch_bounds__(256) void gcn_scatter(
    const float* __restrict__ h, const int* __restrict__ src,
    const int* __restrict__ dst, const float* __restrict__ dinv,
    float* __restrict__ agg)
{
    const int idx = blockIdx.x * blockDim.x + threadIdx.x;  // E*32 threads exact
    const int e = idx >> 5;
    const int p = (idx & 31) * 2;     // feature pair
    const int s = src[e];
    const int d = dst[e];
    const float c = dinv[s] * dinv[d];
    const float2 v = *(const float2*)(h + (size_t)s * GCN_H + p);
    float* out = agg + (size_t)d * GCN_H + p;
    atomicAdd(out + 0, v.x * c);
    atomicAdd(out + 1, v.y * c);
}

// ---------------------------------------------------------------------------
// Epilogue: out = agg + h * dinv^2 + bias, optional relu.  (in-place on agg ok)
// ---------------------------------------------------------------------------
__global__ __launch_bounds__(256) void gcn_epilogue(
    const float* __restrict__ agg, const float* __restrict__ h,
    const float* __restrict__ dinv, const float* __restrict__ bias,
    float* __restrict__ out, int do_relu)
{
    const int idx = blockIdx.x * blockDim.x + threadIdx.x;  // N*64 exact
    const int i = idx >> 6;
    const int f = idx & 63;
    const float di = dinv[i];
    float v = agg[idx] + h[idx] * di * di + bias[f];
    if (do_relu) v = fmaxf(v, 0.0f);
    out[idx] = v;
}

// ---------------------------------------------------------------------------
// Mean pool (accumulate):  psum[batch[i]] += h[i];  pcnt[batch[i]] += 1
// ---------------------------------------------------------------------------
__global__ __launch_bounds__(256) void gcn_pool(
    const float* __restrict__ h, const int* __restrict__ batch,
    float* __restrict__ psum, float* __restrict__ pcnt)
{
    const int idx = blockIdx.x * blockDim.x + threadIdx.x;  // N*64 exact
    const int i = idx >> 6;
    const int f = idx & 63;
    const int b = batch[i];
    atomicAdd(&psum[(size_t)b * GCN_H + f], h[idx]);
    if (f == 0) atomicAdd(&pcnt[b], 1.0f);
}

// ---------------------------------------------------------------------------
// Final linear:  out[b,c] = (psum[b]/max(cnt,1)) @ Wl + bl      (512x64x10)
// ---------------------------------------------------------------------------
__global__ __launch_bounds__(256) void gcn_final(
    const float* __restrict__ psum, const float* __restrict__ pcnt,
    const float* __restrict__ Wl, const float* __restrict__ bl,
    float* __restrict__ out)
{
    const int idx = blockIdx.x * blockDim.x + threadIdx.x;  // B*C = 5120 exact
    const int b = idx / GCN_C;
    const int c = idx % GCN_C;
    const float inv = 1.0f / fmaxf(pcnt[b], 1.0f);
    float acc = bl[c];
    const float* row = psum + (size_t)b * GCN_H;
#pragma unroll 8
    for (int k = 0; k < GCN_H; ++k)
        acc = fmaf(row[k] * inv, Wl[k * GCN_C + c], acc);
    out[idx] = acc;
}

// ---------------------------------------------------------------------------
// Host-side orchestration (graph-capture safe: only async APIs on `stream`)
// ---------------------------------------------------------------------------
extern "C" void kernel_launch(void* const* d_in, const int* in_sizes, int n_in,
                              void* d_out, int out_size, void* d_ws, size_t ws_size,
                              hipStream_t stream) {
    (void)in_sizes; (void)n_in; (void)out_size; (void)ws_size;

    const float* x     = (const float*)d_in[0];
    const int*   ei    = (const int*)  d_in[1];
    const int*   batch = (const int*)  d_in[2];
    const float* W1    = (const float*)d_in[3];
    const float* b1    = (const float*)d_in[4];
    const float* W2    = (const float*)d_in[5];
    const float* b2    = (const float*)d_in[6];
    const float* W3    = (const float*)d_in[7];
    const float* b3    = (const float*)d_in[8];
    const float* Wl    = (const float*)d_in[9];
    const float* bl    = (const float*)d_in[10];
    float* out = (float*)d_out;

    const int* src = ei;           // edge_index[0]
    const int* dst = ei + GCN_E;   // edge_index[1]

    // Workspace layout (floats): dinv[N] | bufA[N*H] | bufB[N*H] | psum[B*H] | pcnt[B]
    float* dinv = (float*)d_ws;
    float* bufA = dinv + GCN_N;
    float* bufB = bufA + (size_t)GCN_N * GCN_H;
    float* psum = bufB + (size_t)GCN_N * GCN_H;
    float* pcnt = psum + (size_t)GCN_B * GCN_H;

    const dim3 gemmBlk(32, 4);
    const dim3 gemmGrd(GCN_N / 16);                 // 3125 row tiles, exact
    const int  scatGrd = (GCN_E * 32) / 256;        // exact
    const int  nhGrd   = (GCN_N * GCN_H) / 256;     // exact
    const size_t nhBytes = (size_t)GCN_N * GCN_H * sizeof(float);

    // Degrees -> dinv
    hipMemsetAsync(dinv, 0, GCN_N * sizeof(float), stream);
    gcn_deg_count<<<(GCN_E + 255) / 256, 256, 0, stream>>>(dst, dinv);
    gcn_dinv<<<(GCN_N + 255) / 256, 256, 0, stream>>>(dinv);

    // ---- Layer 1: h = x@W1 ; agg ; relu(agg + h*dinv^2 + b1) ----
    gcn_gemm_wmma_f32<<<gemmGrd, gemmBlk, 0, stream>>>(x, W1, bufA, GCN_FIN);
    hipMemsetAsync(bufB, 0, nhBytes, stream);
    gcn_scatter<<<scatGrd, 256, 0, stream>>>(bufA, src, dst, dinv, bufB);
    gcn_epilogue<<<nhGrd, 256, 0, stream>>>(bufB, bufA, dinv, b1, bufB, 1);

    // ---- Layer 2 ----
    gcn_gemm_wmma_f32<<<gemmGrd, gemmBlk, 0, stream>>>(bufB, W2, bufA, GCN_H);
    hipMemsetAsync(bufB, 0, nhBytes, stream);
    gcn_scatter<<<scatGrd, 256, 0, stream>>>(bufA, src, dst, dinv, bufB);
    gcn_epilogue<<<nhGrd, 256, 0, stream>>>(bufB, bufA, dinv, b2, bufB, 1);

    // ---- Layer 3 (no relu) ----
    gcn_gemm_wmma_f32<<<gemmGrd, gemmBlk, 0, stream>>>(bufB, W3, bufA, GCN_H);
    hipMemsetAsync(bufB, 0, nhBytes, stream);
    gcn_scatter<<<scatGrd, 256, 0, stream>>>(bufA, src, dst, dinv, bufB);
    gcn_epilogue<<<nhGrd, 256, 0, stream>>>(bufB, bufA, dinv, b3, bufB, 0);

    // ---- Mean pool + final linear ----
    hipMemsetAsync(psum, 0, (size_t)(GCN_B * GCN_H + GCN_B) * sizeof(float), stream);
    gcn_pool<<<nhGrd, 256, 0, stream>>>(bufB, batch, psum, pcnt);
    gcn_final<<<(GCN_B * GCN_C) / 256, 256, 0, stream>>>(psum, pcnt, Wl, bl, out);
}